// LinearAttention_65867618452181
// MI455X (gfx1250) — compile-verified
//
#include <hip/hip_runtime.h>
#include <hip/hip_bf16.h>

typedef __attribute__((ext_vector_type(2))) float v2f;
typedef __attribute__((ext_vector_type(8))) float v8f;

#define BH      64                 // B*H = 4*16
#define S_LEN   4096
#define D_DIM   64
#define SPLIT   8
#define CHUNK   (S_LEN / SPLIT)    // 512 s-rows per block in phase 1
#define SUB     64                 // s-rows staged per LDS pass (32 s-pairs)
#define PP      160                // s-pair pitch (dwords): 160 % 64 == 32 -> disjoint banks for lane halves
#define PQ      68                 // nnq pitch (row-major), 16B aligned, conflict-free A reads

#define POS_SCALE 3.83495197e-4f   // (pi/2) / 4096

__device__ __forceinline__ float non_neg(float x) {
    return x < 0.0f ? __expf(x) : x + 1.0f;
}

// ---------------------------------------------------------------------------
// Phase 1: partial kv_cos / kv_sin per (bh, s-chunk).
// LDS layout: pair-interleaved [s/2][col*2 + (s&1)] so each WMMA operand
// (K = vgpr + 2*(lane/16)) is one aligned ds_load_b64.
// Weight on k: lkc = nn_k*cos, lks = nn_k*sin, lv = v  -> 4 wmma : 4 b64.
// partial layout: [bh][chunk][w=cos/sin][64*64]
// ---------------------------------------------------------------------------
__global__ __launch_bounds__(256) void kv_partial_kernel(
        const float* __restrict__ k, const float* __restrict__ v,
        float* __restrict__ partial) {
    __shared__ float lkc[(SUB / 2) * PP];
    __shared__ float lks[(SUB / 2) * PP];
    __shared__ float lv [(SUB / 2) * PP];

    const int bh    = blockIdx.x / SPLIT;
    const int chunk = blockIdx.x % SPLIT;
    const int tid   = threadIdx.x;
    const int lane  = tid & 31;
    const int wave  = tid >> 5;            // 0..7
    const int half  = lane >> 4;
    const int l16   = lane & 15;

    // tile ownership: dt = wave>>1 (each d-tile by 2 waves), e-tile pair by wave&1
    const int dt   = wave >> 1;
    const int e0   = (wave & 1) * 32;      // col base of first e-tile
    const int e1   = e0 + 16;
    const int dcol = dt * 16 + l16;

    // staging: thread owns s-pair p (rows 2p, 2p+1) and 8 columns
    const int p  = tid >> 3;               // 0..31
    const int cg = tid & 7;                // 0..7

    const float* kb = k + (size_t)bh * S_LEN * D_DIM;
    const float* vb = v + (size_t)bh * S_LEN * D_DIM;

    v8f accC0 = {}, accS0 = {}, accC1 = {}, accS1 = {};

    for (int sub = 0; sub < CHUNK / SUB; ++sub) {
        const int s0 = chunk * CHUNK + sub * SUB;
        // ---- stage 64 rows: nn_k*cos, nn_k*sin, raw v (pair-interleaved) ----
        {
            const int gs0 = s0 + 2 * p;
            float sn0, cs0, sn1, cs1;
            __sincosf(POS_SCALE * (float)gs0, &sn0, &cs0);
            __sincosf(POS_SCALE * (float)(gs0 + 1), &sn1, &cs1);
            const float4* k0 = (const float4*)(kb + (size_t)gs0 * D_DIM);
            const float4* k1 = (const float4*)(kb + (size_t)(gs0 + 1) * D_DIM);
            const float4* v0 = (const float4*)(vb + (size_t)gs0 * D_DIM);
            const float4* v1 = (const float4*)(vb + (size_t)(gs0 + 1) * D_DIM);
#pragma unroll
            for (int j = 0; j < 2; ++j) {
                const int c4 = cg * 2 + j;           // float4 column 0..15
                float4 ka = k0[c4], kbq = k1[c4];
                float4 va = v0[c4], vbq = v1[c4];
                const float* kaf = (const float*)&ka;
                const float* kbf = (const float*)&kbq;
                const float* vaf = (const float*)&va;
                const float* vbf = (const float*)&vbq;
#pragma unroll
                for (int ci = 0; ci < 4; ++ci) {
                    const int c = c4 * 4 + ci;       // column 0..63
                    const float n0 = non_neg(kaf[ci]);
                    const float n1 = non_neg(kbf[ci]);
                    *(v2f*)&lkc[p * PP + c * 2] = (v2f){ n0 * cs0, n1 * cs1 };
                    *(v2f*)&lks[p * PP + c * 2] = (v2f){ n0 * sn0, n1 * sn1 };
                    *(v2f*)&lv [p * PP + c * 2] = (v2f){ vaf[ci],  vbf[ci]  };
                }
            }
        }
        __syncthreads();
        // ---- accumulate: 4 wmma per 4 ds_load_b64 per K-step of 4 ----
#pragma unroll
        for (int ss = 0; ss < SUB; ss += 4) {
            const int pp = (ss >> 1) + half;         // s-pair index for this lane half
            v2f Ac = *(const v2f*)&lkc[pp * PP + dcol * 2];
            v2f As = *(const v2f*)&lks[pp * PP + dcol * 2];
            v2f B0 = *(const v2f*)&lv [pp * PP + (e0 + l16) * 2];
            v2f B1 = *(const v2f*)&lv [pp * PP + (e1 + l16) * 2];
            accC0 = __builtin_amdgcn_wmma_f32_16x16x4_f32(false, Ac, false, B0, (short)0, accC0, false, false);
            accS0 = __builtin_amdgcn_wmma_f32_16x16x4_f32(false, As, false, B0, (short)0, accS0, false, false);
            accC1 = __builtin_amdgcn_wmma_f32_16x16x4_f32(false, Ac, false, B1, (short)0, accC1, false, false);
            accS1 = __builtin_amdgcn_wmma_f32_16x16x4_f32(false, As, false, B1, (short)0, accS1, false, false);
        }
        __syncthreads();
    }

    // ---- store partials (C/D: lane = col N, vgpr m -> row M = m + 8*half) ----
    float* pc_ = partial + ((size_t)(bh * SPLIT + chunk) * 2 + 0) * (D_DIM * D_DIM);
    float* ps_ = partial + ((size_t)(bh * SPLIT + chunk) * 2 + 1) * (D_DIM * D_DIM);
#pragma unroll
    for (int m = 0; m < 8; ++m) {
        const int drow = dt * 16 + m + 8 * half;
        pc_[drow * D_DIM + e0 + l16] = accC0[m];
        ps_[drow * D_DIM + e0 + l16] = accS0[m];
        pc_[drow * D_DIM + e1 + l16] = accC1[m];
        ps_[drow * D_DIM + e1 + l16] = accS1[m];
    }
}

// ---------------------------------------------------------------------------
// Phase 2: reduce the SPLIT partials (fixed order -> deterministic)
// kv layout: [bh][w][64*64]
// ---------------------------------------------------------------------------
__global__ __launch_bounds__(256) void kv_reduce_kernel(
        const float* __restrict__ partial, float* __restrict__ kv) {
    const int idx  = blockIdx.x * 256 + threadIdx.x;   // < 64*2*4096
    const int bh   = idx >> 13;
    const int rest = idx & 8191;                       // w*4096 + d*64 + e
    float acc = 0.0f;
#pragma unroll
    for (int c = 0; c < SPLIT; ++c)
        acc += partial[((size_t)(bh * SPLIT + c) * 2) * 4096 + rest];
    kv[idx] = acc;
}

// ---------------------------------------------------------------------------
// Phase 3: out[s,:] = cos(s)*(qf@kv_cos)[s,:] + sin(s)*(qf@kv_sin)[s,:]
// cos/sin hoisted out of the K-loop: inner loop is 2 wmma : 3 ds_load_b64.
// kv in LDS pair-interleaved along d; nnq row-major (pairs along d contiguous).
// ---------------------------------------------------------------------------
__global__ __launch_bounds__(256) void out_kernel(
        const float* __restrict__ q, const float* __restrict__ kvbuf,
        float* __restrict__ out) {
    __shared__ float kvcL[(D_DIM / 2) * PP];
    __shared__ float kvsL[(D_DIM / 2) * PP];
    __shared__ float nnq [SUB * PQ];         // 64 q rows, non_neg applied

    const int nblk  = S_LEN / 64;
    const int bh    = blockIdx.x / nblk;
    const int rb    = blockIdx.x % nblk;
    const int sbase = rb * 64;
    const int tid   = threadIdx.x;
    const int lane  = tid & 31;
    const int wave  = tid >> 5;
    const int half  = lane >> 4;
    const int l16   = lane & 15;

    // ---- stage kv_cos / kv_sin pair-interleaved along d ----
    {
        const float* kvcg = kvbuf + (size_t)(bh * 2) * 4096;
        const float* kvsg = kvcg + 4096;
        const int p  = tid >> 3;             // d-pair 0..31
        const int cg = tid & 7;
#pragma unroll
        for (int j = 0; j < 2; ++j) {
            const int c4 = cg * 2 + j;
            float4 c0 = ((const float4*)(kvcg + (size_t)(2 * p) * D_DIM))[c4];
            float4 c1 = ((const float4*)(kvcg + (size_t)(2 * p + 1) * D_DIM))[c4];
            float4 s0 = ((const float4*)(kvsg + (size_t)(2 * p) * D_DIM))[c4];
            float4 s1 = ((const float4*)(kvsg + (size_t)(2 * p + 1) * D_DIM))[c4];
            const float* c0f = (const float*)&c0;
            const float* c1f = (const float*)&c1;
            const float* s0f = (const float*)&s0;
            const float* s1f = (const float*)&s1;
#pragma unroll
            for (int ci = 0; ci < 4; ++ci) {
                const int e = c4 * 4 + ci;
                *(v2f*)&kvcL[p * PP + e * 2] = (v2f){ c0f[ci], c1f[ci] };
                *(v2f*)&kvsL[p * PP + e * 2] = (v2f){ s0f[ci], s1f[ci] };
            }
        }
    }
    // ---- stage 64 rows of non_neg(q), row-major ----
    {
        const int row = tid >> 2, seg = tid & 3;
        const float4* qrow =
            (const float4*)(q + ((size_t)bh * S_LEN + sbase + row) * D_DIM);
#pragma unroll
        for (int j = 0; j < 4; ++j) {
            const int c4 = seg + 4 * j;
            float4 qq = qrow[c4];
            float4 nn;
            nn.x = non_neg(qq.x); nn.y = non_neg(qq.y);
            nn.z = non_neg(qq.z); nn.w = non_neg(qq.w);
            *(float4*)&nnq[row * PQ + c4 * 4] = nn;
        }
    }
    __syncthreads();

    const int rt    = wave >> 1;             // row tile 0..3
    const int r0    = rt * 16;
    const int ebase = (wave & 1) * 2;        // each wave: 2 of 4 e-tiles

    // per-output-row cos/sin (row M = m + 8*half in the D layout)
    float cm[8], sm[8];
#pragma unroll
    for (int m = 0; m < 8; ++m) {
        const int sr = sbase + r0 + m + 8 * half;
        __sincosf(POS_SCALE * (float)sr, &sm[m], &cm[m]);
    }

    float* ob = out + (size_t)bh * S_LEN * D_DIM;

    for (int ei = 0; ei < 2; ++ei) {
        const int e = (ebase + ei) * 16;
        v8f accX = {}, accY = {};
#pragma unroll
        for (int kk = 0; kk < D_DIM; kk += 4) {
            v2f A  = *(const v2f*)&nnq[(r0 + l16) * PQ + kk + 2 * half];
            const int dp = (kk >> 1) + half;        // d-pair index
            v2f Bc = *(const v2f*)&kvcL[dp * PP + (e + l16) * 2];
            v2f Bs = *(const v2f*)&kvsL[dp * PP + (e + l16) * 2];
            accX = __builtin_amdgcn_wmma_f32_16x16x4_f32(false, A, false, Bc, (short)0, accX, false, false);
            accY = __builtin_amdgcn_wmma_f32_16x16x4_f32(false, A, false, Bs, (short)0, accY, false, false);
        }
#pragma unroll
        for (int m = 0; m < 8; ++m) {
            const int sr = sbase + r0 + m + 8 * half;
            ob[(size_t)sr * D_DIM + e + l16] = cm[m] * accX[m] + sm[m] * accY[m];
        }
    }
}

// ---------------------------------------------------------------------------
extern "C" void kernel_launch(void* const* d_in, const int* in_sizes, int n_in,
                              void* d_out, int out_size, void* d_ws, size_t ws_size,
                              hipStream_t stream) {
    (void)in_sizes; (void)n_in; (void)out_size; (void)ws_size;
    const float* q = (const float*)d_in[0];
    const float* k = (const float*)d_in[1];
    const float* v = (const float*)d_in[2];
    float* out = (float*)d_out;

    float* partial = (float*)d_ws;                               // 16 MB
    float* kv      = partial + (size_t)BH * SPLIT * 2 * 4096;    // 2 MB

    kv_partial_kernel<<<BH * SPLIT, 256, 0, stream>>>(k, v, partial);
    kv_reduce_kernel<<<(BH * 2 * 4096) / 256, 256, 0, stream>>>(partial, kv);
    out_kernel<<<BH * (S_LEN / 64), 256, 0, stream>>>(q, kv, out);
}